// GraphSToV_6451040878944
// MI455X (gfx1250) — compile-verified
//
#include <hip/hip_runtime.h>

// Problem constants from the reference: B=16, N=128, F=64, K=128, C=3
#define B_ 16
#define N_ 128
#define F_ 64
#define K_ 128
#define C_ 3

typedef __attribute__((ext_vector_type(2))) float v2f;
typedef __attribute__((ext_vector_type(4))) float v4f;
typedef __attribute__((ext_vector_type(8))) float v8f;

// ---------------------------------------------------------------------------
// Kernel 1: pa[b,i,k] = sum_f sf[b,i,f]*w[f,k]
//           pb[b,j,k] = sum_f sf[b,j,f]*w[F+f,k]
// One wave per 16x16 (row,k) tile; both products share the A fragment.
// Uses V_WMMA_F32_16X16X4_F32 (full f32 precision; GEMM is ~67 MFLOP, trivial).
//
// Fragment layouts (ISA 7.12.2, wave32):
//   A (16x4 f32, 2 VGPRs): lane L<16 -> M=L, K={kk,kk+1}; lane L>=16 -> M=L-16, K={kk+2,kk+3}
//   B (4x16 f32, 2 VGPRs): lane L<16 -> N=L, K={kk,kk+1}; lane L>=16 -> N=L-16, K={kk+2,kk+3}
//   C/D (16x16 f32, 8 VGPRs): VGPR r: lanes 0-15 = (M=r, N=lane), lanes 16-31 = (M=r+8, N=lane-16)
// ---------------------------------------------------------------------------
__global__ __launch_bounds__(256) void pa_pb_gemm_wmma(
    const float* __restrict__ sf,   // (B,N,F)
    const float* __restrict__ w,    // (2F,K)
    float* __restrict__ pa,         // (B,N,K)
    float* __restrict__ pb)         // (B,N,K)
{
  const int wave = blockIdx.x * (blockDim.x >> 5) + (threadIdx.x >> 5);
  const int lane = threadIdx.x & 31;

  // tile decomposition: wave -> (b, row-tile, k-tile); 8 k-tiles, 8 row-tiles, 16 batches
  const int kt = wave & 7;
  const int it = (wave >> 3) & 7;
  const int b  = wave >> 6;

  const int laneM = lane & 15;   // A row / B,D column within tile
  const int laneH = lane >> 4;   // lane half selects K sub-pair (A/B) or M+8 (D)

  const float* sfB = sf + (size_t)b * N_ * F_;
  const int row = it * 16 + laneM;   // A-matrix row this lane feeds
  const int col = kt * 16 + laneM;   // B/D column this lane feeds

  v8f accA = {};
  v8f accB = {};

  for (int kk = 0; kk < F_; kk += 4) {
    const int ka = kk + laneH * 2;
    v2f a;
    a.x = sfB[row * F_ + ka];
    a.y = sfB[row * F_ + ka + 1];
    v2f bA, bB;
    bA.x = w[(ka    ) * K_ + col];
    bA.y = w[(ka + 1) * K_ + col];
    bB.x = w[(F_ + ka    ) * K_ + col];
    bB.y = w[(F_ + ka + 1) * K_ + col];
    // (neg_a, A, neg_b, B, c_mod, C, reuse_a, reuse_b)
    accA = __builtin_amdgcn_wmma_f32_16x16x4_f32(false, a, false, bA, (short)0, accA, false, false);
    accB = __builtin_amdgcn_wmma_f32_16x16x4_f32(false, a, false, bB, (short)0, accB, false, false);
  }

  float* paB = pa + ((size_t)b * N_ + it * 16) * K_ + kt * 16;
  float* pbB = pb + ((size_t)b * N_ + it * 16) * K_ + kt * 16;
#pragma unroll
  for (int r = 0; r < 8; ++r) {
    const int rr = laneH * 8 + r;            // D-matrix row for this lane half
    paB[rr * K_ + laneM] = accA[r];
    pbB[rr * K_ + laneM] = accB[r];
  }
}

// ---------------------------------------------------------------------------
// Kernel 2: out[b,i,j,c,k] = (pa[b,i,k] + pb[b,j,k] + bsv[k]) * dist[b,i,j,c]
// One wave per (b,i,j); lane owns k = 4*lane..4*lane+3 (32*4 = K = 128).
// Output is 402 MB (>> 192 MB L2) -> non-temporal 128-bit stores.
// pa/pb/bias/dist reads are small + heavily reused -> default (RT) caching.
// ---------------------------------------------------------------------------
__global__ __launch_bounds__(256) void expand_mul(
    const float* __restrict__ pa,    // (B,N,K)
    const float* __restrict__ pb,    // (B,N,K)
    const float* __restrict__ bsv,   // (K)
    const float* __restrict__ dist,  // (B,N,N,C)
    float* __restrict__ out)         // (B,N,N,C,K)
{
  const int wid  = blockIdx.x * (blockDim.x >> 5) + (threadIdx.x >> 5);
  const int lane = threadIdx.x & 31;

  const int j = wid & (N_ - 1);
  const int i = (wid >> 7) & (N_ - 1);
  const int b = wid >> 14;

  const int k4 = lane * 4;
  const v4f va = *(const v4f*)(pa  + ((size_t)(b * N_ + i)) * K_ + k4);
  const v4f vb = *(const v4f*)(pb  + ((size_t)(b * N_ + j)) * K_ + k4);
  const v4f bb = *(const v4f*)(bsv + k4);

  v4f s = va + vb + bb;

  const size_t dIdx = (((size_t)b * N_ + i) * N_ + j) * C_;
  float* ob = out + dIdx * K_ + k4;

#pragma unroll
  for (int c = 0; c < C_; ++c) {
    const float d = dist[dIdx + c];        // wave-uniform scalar
    v4f o = s * d;
    __builtin_nontemporal_store(o, (v4f*)(ob + (size_t)c * K_));
  }
}

// ---------------------------------------------------------------------------
extern "C" void kernel_launch(void* const* d_in, const int* in_sizes, int n_in,
                              void* d_out, int out_size, void* d_ws, size_t ws_size,
                              hipStream_t stream) {
  (void)in_sizes; (void)n_in; (void)out_size; (void)ws_size;

  const float* sf   = (const float*)d_in[0];  // scalar_features (16,128,64)
  const float* dist = (const float*)d_in[1];  // distances (16,128,128,3)
  const float* w    = (const float*)d_in[2];  // w_sv (128,128)
  const float* bsv  = (const float*)d_in[3];  // b_sv (128,)
  float* out = (float*)d_out;                 // (16,128,128,3,128)

  float* pa = (float*)d_ws;                   // (16,128,128) = 1 MB
  float* pb = pa + (size_t)B_ * N_ * K_;      // (16,128,128) = 1 MB

  // 1024 tile-waves / 8 waves per 256-thread block = 128 blocks
  pa_pb_gemm_wmma<<<(B_ * 8 * 8) / 8, 256, 0, stream>>>(sf, w, pa, pb);

  // 262144 (b,i,j) waves / 8 per block = 32768 blocks
  expand_mul<<<(B_ * N_ * N_) / 8, 256, 0, stream>>>(pa, pb, bsv, dist, out);
}